// Linear_12051678233049
// MI455X (gfx1250) — compile-verified
//
#include <hip/hip_runtime.h>
#include <math.h>

// ---------------------------------------------------------------------------
// Bayesian linear layer (reparameterized weights), MI455X / gfx1250.
//   out[b,o] = sum_i x[b,i]*(mu_w[o,i] + softplus(rho_w[o,i])*eps_w[b,o,i])
//            + mu_b[o] + softplus(rho_b[o])*eps_b[b,o]
// Memory-bound on the 512 MiB eps_w stream (~22us @ 23.3 TB/s).
// Split: fp32 WMMA GEMM for x@mu_w^T (+bias epilogue), NT-streamed vector-FMA
// reduction for the eps term, softplus(rho_w) precomputed once into d_ws.
// ---------------------------------------------------------------------------

typedef __attribute__((ext_vector_type(2))) float v2f;
typedef __attribute__((ext_vector_type(4))) float v4f;
typedef __attribute__((ext_vector_type(8))) float v8f;

#define B_DIM   128
#define IN_DIM  1024
#define OUT_DIM 1024

__device__ __forceinline__ float softplus_f(float v) {
    // numerically stable: max(v,0) + log1p(exp(-|v|))
    return fmaxf(v, 0.0f) + log1pf(expf(-fabsf(v)));
}

// ---------------------------------------------------------------------------
// Kernel 0: t[o,i] = softplus(rho_w[o,i])   (1M elems -> 4 MiB workspace)
// ---------------------------------------------------------------------------
__global__ __launch_bounds__(256)
void softplus_precompute_kernel(const float* __restrict__ rho,
                                float* __restrict__ t, int n) {
    int i = blockIdx.x * blockDim.x + threadIdx.x;
    if (i < n) t[i] = softplus_f(rho[i]);
}

// ---------------------------------------------------------------------------
// Kernel 1: out[b,o] = (x @ mu_w^T)[b,o] + mu_b[o] + softplus(rho_b[o])*eps_b
// One wave per 16x16 output tile; K=1024 swept with V_WMMA_F32_16X16X4_F32
// (exact fp32). A layout: lanes 0-15 rows M, VGPR{0,1}=K{0,1}; lanes 16-31
// hold K{2,3}. B mirrored (B[k,n] = mu_w[n*IN + k], contiguous in k).
// ---------------------------------------------------------------------------
__global__ __launch_bounds__(32)
void wmma_mu_bias_kernel(const float* __restrict__ x,
                         const float* __restrict__ wmu,
                         const float* __restrict__ bmu,
                         const float* __restrict__ brho,
                         const float* __restrict__ eps_b,
                         float* __restrict__ out) {
    const int lane  = threadIdx.x;        // 0..31, full wave, EXEC all ones
    const int m0    = blockIdx.y * 16;    // batch-row tile
    const int n0    = blockIdx.x * 16;    // out-col tile
    const int l15   = lane & 15;
    const int khalf = (lane >> 4) * 2;    // 0 for lanes 0-15, 2 for 16-31

    const float* __restrict__ arow = x   + (size_t)(m0 + l15) * IN_DIM + khalf;
    const float* __restrict__ bcol = wmu + (size_t)(n0 + l15) * IN_DIM + khalf;

    v8f c = {};
#pragma unroll 8
    for (int k = 0; k < IN_DIM; k += 4) {
        v2f a = *(const v2f*)(arow + k);   // A[m, k..k+1] (or k+2..k+3)
        v2f b = *(const v2f*)(bcol + k);   // B[k..k+1, n]
        c = __builtin_amdgcn_wmma_f32_16x16x4_f32(
                /*neg_a=*/false, a, /*neg_b=*/false, b,
                /*c_mod=*/(short)0, c, /*reuse_a=*/false, /*reuse_b=*/false);
    }

    // Epilogue: fuse per-sample bias. C/D layout: VGPR r, lanes 0-15 -> M=r,
    // lanes 16-31 -> M=r+8; N = n0 + (lane&15).
    const int   col    = n0 + l15;
    const float bbase  = bmu[col];
    const float sp_b   = softplus_f(brho[col]);
    const int   rowoff = m0 + ((lane >> 4) << 3);
#pragma unroll
    for (int r = 0; r < 8; ++r) {
        const int row = rowoff + r;
        out[(size_t)row * OUT_DIM + col] =
            c[r] + bbase + sp_b * eps_b[(size_t)row * OUT_DIM + col];
    }
}

// ---------------------------------------------------------------------------
// Kernel 2: out[b,o] += sum_i x[b,i] * t[o,i] * eps_w[b,o,i]
// One wave per (b,o). Lanes stream the 4 KiB eps row as float4 (contiguous
// 512 B per wave-issue), non-temporal so the 512 MiB stream bypasses cache
// residency; x (512 KiB) and t (4 MiB) stay hot in L2. shfl_xor reduce (w32).
// ---------------------------------------------------------------------------
__global__ __launch_bounds__(256)
void eps_stream_kernel(const float* __restrict__ x,
                       const float* __restrict__ wrho,
                       const float* __restrict__ t_ws,
                       const float* __restrict__ eps_w,
                       float* __restrict__ out, int use_ws) {
    const int wave = blockIdx.x * (blockDim.x >> 5) + (threadIdx.x >> 5);
    const int lane = threadIdx.x & 31;
    const int b = wave >> 10;             // wave / OUT_DIM
    const int o = wave & (OUT_DIM - 1);   // wave % OUT_DIM

    const float* __restrict__ xr = x    + (size_t)b * IN_DIM;
    const float* __restrict__ tr = t_ws + (size_t)o * IN_DIM;
    const float* __restrict__ rr = wrho + (size_t)o * IN_DIM;
    const float* __restrict__ er =
        eps_w + ((size_t)b * OUT_DIM + (size_t)o) * IN_DIM;

    float acc = 0.0f;
#pragma unroll
    for (int it = 0; it < IN_DIM / 128; ++it) {
        const int i = it * 128 + lane * 4;
        v4f xv = *(const v4f*)(xr + i);
        v4f ev = __builtin_nontemporal_load((const v4f*)(er + i));
        v4f tv;
        if (use_ws) {
            tv = *(const v4f*)(tr + i);
        } else {
            v4f rv = *(const v4f*)(rr + i);
            tv.x = softplus_f(rv.x);
            tv.y = softplus_f(rv.y);
            tv.z = softplus_f(rv.z);
            tv.w = softplus_f(rv.w);
        }
        v4f p = xv * tv * ev;
        acc += (p.x + p.y) + (p.z + p.w);
    }

    // wave32 butterfly reduction
#pragma unroll
    for (int off = 16; off > 0; off >>= 1)
        acc += __shfl_xor(acc, off, 32);

    if (lane == 0)
        out[wave] += acc;   // wave == b*OUT_DIM + o; single writer, ordered
}

// ---------------------------------------------------------------------------
extern "C" void kernel_launch(void* const* d_in, const int* in_sizes, int n_in,
                              void* d_out, int out_size, void* d_ws, size_t ws_size,
                              hipStream_t stream) {
    const float* x     = (const float*)d_in[0];  // (128, 1024)
    const float* wmu   = (const float*)d_in[1];  // (1024, 1024)
    const float* wrho  = (const float*)d_in[2];  // (1024, 1024)
    const float* bmu   = (const float*)d_in[3];  // (1024,)
    const float* brho  = (const float*)d_in[4];  // (1024,)
    const float* eps_w = (const float*)d_in[5];  // (128, 1024, 1024)
    const float* eps_b = (const float*)d_in[6];  // (128, 1024)
    float*       out   = (float*)d_out;          // (128, 1024)
    float*       t     = (float*)d_ws;           // softplus(rho_w), 4 MiB

    const int nW = IN_DIM * OUT_DIM;
    const int use_ws = (ws_size >= (size_t)nW * sizeof(float)) ? 1 : 0;

    if (use_ws) {
        softplus_precompute_kernel<<<(nW + 255) / 256, 256, 0, stream>>>(wrho, t, nW);
    }

    dim3 gridA(OUT_DIM / 16, B_DIM / 16);       // 64 x 8 tiles, 1 wave each
    wmma_mu_bias_kernel<<<gridA, 32, 0, stream>>>(x, wmu, bmu, brho, eps_b, out);

    const int total_waves = B_DIM * OUT_DIM;    // one wave per output element
    eps_stream_kernel<<<total_waves / 8, 256, 0, stream>>>(x, wrho, t, eps_w,
                                                           out, use_ws);
}